// GNN_50491635531994
// MI455X (gfx1250) — compile-verified
//
#include <hip/hip_runtime.h>

#define N_NODES 50000
#define N_EDGES 800000
#define IN_DIM  128
#define HID_DIM 256
#define OUT_DIM 128

typedef float v2f __attribute__((ext_vector_type(2)));
typedef float v8f __attribute__((ext_vector_type(8)));

// ---------------------------------------------------------------------------
// Utility kernels
// ---------------------------------------------------------------------------
__global__ void fill_f32(float* __restrict__ p, float v, int n) {
    int i = blockIdx.x * blockDim.x + threadIdx.x;
    if (i < n) p[i] = v;
}

__global__ void count_deg(const long long* __restrict__ dst,
                          float* __restrict__ deg, int E) {
    int e = blockIdx.x * blockDim.x + threadIdx.x;
    if (e < E) atomicAdd(&deg[(int)dst[e]], 1.0f);
}

__global__ void rsqrt_ip(float* __restrict__ p, int n) {
    int i = blockIdx.x * blockDim.x + threadIdx.x;
    if (i < n) p[i] = rsqrtf(p[i]);
}

// ---------------------------------------------------------------------------
// fp32 WMMA GEMM: C[M,N] = A[M,K] * B[K,N], row-major, exact fp32 via
// V_WMMA_F32_16X16X4_F32.
//
// Block = 256 threads (8 waves). blockIdx.y selects a 64-column strip of B,
// which is staged ONCE into LDS transposed (stride K+4: 8B-aligned ds_load_b64,
// conflict-free reads). Each wave computes a 16x64 output strip (4 WMMA tiles)
// for its own M-tile, so one A global_load_b64 feeds 4 WMMAs and every B
// operand is a single ds_load_b64 with immediate offsets (K, N templated).
// All WMMA-guarding branches are wave-uniform -> EXEC stays all-1s.
// ---------------------------------------------------------------------------
template<int K, int N>
__global__ __launch_bounds__(256)
void gemm_wmma_f32(const float* __restrict__ A, const float* __restrict__ B,
                   float* __restrict__ C, int M) {
    constexpr int KP = K + 4;          // pad: keeps 8B alignment, spreads banks
    __shared__ float Bs[64 * KP];

    const int lane = threadIdx.x & 31;
    const int wave = threadIdx.x >> 5;
    const int half = lane >> 4;        // selects K-pair {0,1} vs {2,3}
    const int l16  = lane & 15;
    const int col0 = blockIdx.y * 64;

    // ---- stage B[0:K, col0:col0+64] into LDS, transposed ----
    {
        const int c  = threadIdx.x & 63;   // column within strip (coalesced)
        const int k0 = threadIdx.x >> 6;   // 0..3
#pragma unroll
        for (int k = k0; k < K; k += 4)
            Bs[c * KP + k] = B[(size_t)k * N + col0 + c];
    }
    __syncthreads();

    const int row0 = (blockIdx.x * 8 + wave) * 16;
    if (row0 < M) {
        // A 16x4 layout: lanes 0-15 rows hold K=k+0,k+1; lanes 16-31 K=k+2,k+3
        const float* Arow = A + (size_t)(row0 + l16) * K + 2 * half;
        // B^T in LDS: column (l16 + 16*strip), contiguous K
        const float* Bl = Bs + l16 * KP + 2 * half;

        v8f acc0 = {}, acc1 = {}, acc2 = {}, acc3 = {};
#pragma unroll
        for (int k = 0; k < K; k += 4) {
            v2f a  = *(const v2f*)(Arow + k);
            v2f b0 = *(const v2f*)(Bl + 0 * 16 * KP + k);
            v2f b1 = *(const v2f*)(Bl + 1 * 16 * KP + k);
            v2f b2 = *(const v2f*)(Bl + 2 * 16 * KP + k);
            v2f b3 = *(const v2f*)(Bl + 3 * 16 * KP + k);
            acc0 = __builtin_amdgcn_wmma_f32_16x16x4_f32(false, a, false, b0,
                                                         (short)0, acc0, false, false);
            acc1 = __builtin_amdgcn_wmma_f32_16x16x4_f32(false, a, false, b1,
                                                         (short)0, acc1, false, false);
            acc2 = __builtin_amdgcn_wmma_f32_16x16x4_f32(false, a, false, b2,
                                                         (short)0, acc2, false, false);
            acc3 = __builtin_amdgcn_wmma_f32_16x16x4_f32(false, a, false, b3,
                                                         (short)0, acc3, false, false);
        }

        // D 16x16 layout: VGPR v -> row v + 8*half, col = l16 (+16 per strip)
        float* Crow = C + (size_t)(row0 + 8 * half) * N + col0 + l16;
#pragma unroll
        for (int v = 0; v < 8; ++v) {
            Crow[(size_t)v * N +  0] = acc0[v];
            Crow[(size_t)v * N + 16] = acc1[v];
            Crow[(size_t)v * N + 32] = acc2[v];
            Crow[(size_t)v * N + 48] = acc3[v];
        }
    }
}

// ---------------------------------------------------------------------------
// Edge aggregation: one wave per edge; float4 gathers (global_load_b128) from
// h[src], scalar f32 atomics into agg[dst]. h and agg are L2-resident
// (<= 51 MB << 192 MB L2), so the random access stays on-die.
// ---------------------------------------------------------------------------
template<int F>
__global__ __launch_bounds__(256)
void edge_agg(const float* __restrict__ h, const float* __restrict__ dinv,
              const long long* __restrict__ src, const long long* __restrict__ dst,
              float* __restrict__ agg, int E) {
    int e = blockIdx.x * 8 + (threadIdx.x >> 5);
    if (e >= E) return;
    const int lane = threadIdx.x & 31;
    const int s = (int)src[e];
    const int d = (int)dst[e];
    const float norm = dinv[s] * dinv[d];
    const float4* hs = (const float4*)(h + (size_t)s * F);
    float* ad = agg + (size_t)d * F;
#pragma unroll
    for (int i = 0; i < F / 128; ++i) {
        float4 v = hs[lane + 32 * i];
        const int base = (lane + 32 * i) * 4;
        atomicAdd(&ad[base + 0], v.x * norm);
        atomicAdd(&ad[base + 1], v.y * norm);
        atomicAdd(&ad[base + 2], v.z * norm);
        atomicAdd(&ad[base + 3], v.w * norm);
    }
}

// ---------------------------------------------------------------------------
// Self-loop + bias (+ optional ReLU), in place on agg. F templated -> shifts.
// ---------------------------------------------------------------------------
template<bool RELU, int F>
__global__ void finalize(float* __restrict__ agg, const float* __restrict__ h,
                         const float* __restrict__ dinv, const float* __restrict__ b,
                         int total) {
    int idx = blockIdx.x * blockDim.x + threadIdx.x;
    if (idx >= total) return;
    const int node = idx / F;          // F is a power of two -> shift
    const int f = idx & (F - 1);
    const float di = dinv[node];
    float v = agg[idx] + h[idx] * (di * di) + b[f];
    agg[idx] = RELU ? fmaxf(v, 0.0f) : v;
}

// ---------------------------------------------------------------------------
// Orchestration
// ---------------------------------------------------------------------------
extern "C" void kernel_launch(void* const* d_in, const int* in_sizes, int n_in,
                              void* d_out, int out_size, void* d_ws, size_t ws_size,
                              hipStream_t stream) {
    const float*     x   = (const float*)d_in[0];
    const long long* ei  = (const long long*)d_in[1];   // int64 [2, E]
    const float*     W1  = (const float*)d_in[2];
    const float*     b1  = (const float*)d_in[3];
    const float*     W2  = (const float*)d_in[4];
    const float*     b2  = (const float*)d_in[5];
    const long long* src = ei;
    const long long* dst = ei + N_EDGES;
    float* out = (float*)d_out;

    // Workspace carve-up (~128.2 MB)
    float* dinv = (float*)d_ws;                                // N
    float* h1   = dinv + N_NODES;                              // N*HID
    float* a1   = h1 + (size_t)N_NODES * HID_DIM;              // N*HID
    float* h2   = a1 + (size_t)N_NODES * HID_DIM;              // N*OUT

    const int T = 256;
    auto blk = [](long long n, int t) { return (unsigned)((n + t - 1) / t); };

    // --- degrees: deg = 1 + #incoming; dinv = rsqrt(deg) ---
    fill_f32 <<<blk(N_NODES, T), T, 0, stream>>>(dinv, 1.0f, N_NODES);
    count_deg<<<blk(N_EDGES, T), T, 0, stream>>>(dst, dinv, N_EDGES);
    rsqrt_ip <<<blk(N_NODES, T), T, 0, stream>>>(dinv, N_NODES);

    const unsigned mblk = blk(N_NODES / 16, 8);   // 8 M-tiles per block

    // --- layer 1: h1 = x @ W1 ---
    gemm_wmma_f32<IN_DIM, HID_DIM>
        <<<dim3(mblk, HID_DIM / 64), T, 0, stream>>>(x, W1, h1, N_NODES);

    fill_f32<<<blk((long long)N_NODES * HID_DIM, T), T, 0, stream>>>(
        a1, 0.0f, N_NODES * HID_DIM);
    edge_agg<HID_DIM><<<blk(N_EDGES, 8), T, 0, stream>>>(
        h1, dinv, src, dst, a1, N_EDGES);
    finalize<true, HID_DIM><<<blk((long long)N_NODES * HID_DIM, T), T, 0, stream>>>(
        a1, h1, dinv, b1, N_NODES * HID_DIM);

    // --- layer 2: h2 = relu_out @ W2 ---
    gemm_wmma_f32<HID_DIM, OUT_DIM>
        <<<dim3(mblk, OUT_DIM / 64), T, 0, stream>>>(a1, W2, h2, N_NODES);

    fill_f32<<<blk((long long)N_NODES * OUT_DIM, T), T, 0, stream>>>(
        out, 0.0f, N_NODES * OUT_DIM);
    edge_agg<OUT_DIM><<<blk(N_EDGES, 8), T, 0, stream>>>(
        h2, dinv, src, dst, out, N_EDGES);
    finalize<false, OUT_DIM><<<blk((long long)N_NODES * OUT_DIM, T), T, 0, stream>>>(
        out, h2, dinv, b2, N_NODES * OUT_DIM);
}